// GraphAttention_57251914055966
// MI455X (gfx1250) — compile-verified
//
#include <hip/hip_runtime.h>
#include <hip/hip_bf16.h>

#define B 4
#define F 64
#define N 8192
#define SLOPE 0.2f

typedef float v2f __attribute__((ext_vector_type(2)));
typedef float v4f __attribute__((ext_vector_type(4)));
typedef float v8f __attribute__((ext_vector_type(8)));

// ---------------------------------------------------------------------------
// Kernel 1: e[b,o,n] = sum_f W[o,f] * h[b,f,n] via V_WMMA_F32_16X16X4_F32.
// One wave per 16-column tile of e. A = W padded to 16 rows (rows 2..15 = 0),
// B = 4x16 slice of h, 16 chained WMMAs cover K = 64.
//
// A rows 2..15 are zeroed with a branch-free select on a clamped-address load
// (avoids divergent saveexec/cbranch codegen around predicated loads).
// ---------------------------------------------------------------------------
__global__ void __launch_bounds__(256)
gat_proj_wmma(const float* __restrict__ h, const float* __restrict__ W,
              float* __restrict__ e1, float* __restrict__ e2) {
    const int lane = threadIdx.x & 31;
    const int wave = (blockIdx.x * (blockDim.x >> 5)) + (threadIdx.x >> 5);
    const int tilesPerBatch = N / 16;                 // 512
    const int b  = wave / tilesPerBatch;              // 0..3
    const int n0 = (wave % tilesPerBatch) * 16;       // column tile base

    const int  hi   = lane >> 4;                      // 0 or 1 (K sub-pair)
    const int  m    = lane & 15;                      // A row
    const int  col  = lane & 15;                      // B col
    const bool real = (m < 2);                        // rows 0,1 carry W

    // Clamped, always-valid W row pointer; rows >=2 read row 0 then select 0.
    const float* __restrict__ wrow = W + (real ? m : 0) * F;
    const float* __restrict__ hb   = h + (size_t)b * F * N;

    v8f c = {};
    #pragma unroll
    for (int k0 = 0; k0 < F; k0 += 4) {
        const int ka = k0 + 2 * hi;
        const float w0 = wrow[ka + 0];                // unconditional load
        const float w1 = wrow[ka + 1];
        v2f a, bm;
        a[0] = real ? w0 : 0.0f;                      // v_cndmask, no branch
        a[1] = real ? w1 : 0.0f;
        bm[0] = hb[(size_t)(ka + 0) * N + n0 + col];
        bm[1] = hb[(size_t)(ka + 1) * N + n0 + col];
        c = __builtin_amdgcn_wmma_f32_16x16x4_f32(
                /*neg_a=*/false, a, /*neg_b=*/false, bm,
                /*c_mod=*/(short)0, c, /*reuse_a=*/false, /*reuse_b=*/false);
    }

    if (lane < 16) {                                  // rows 0,1 of D
        e1[(size_t)b * N + n0 + lane] = c[0];
        e2[(size_t)b * N + n0 + lane] = c[1];
    }
}

// ---------------------------------------------------------------------------
// Kernel 2: fused lrelu + adjacency mask + row softmax.
// One block per (b, i) row. Row of 8192 masked scores lives in LDS (32 KB):
//   pass A: compute s = lrelu(e1[i]+e2[j]) with mask -> LDS, local max
//   reduce max; pass B: exp(s-m) -> LDS, local sum; reduce sum
//   pass C: scaled NT write of 1 GiB output (never re-read -> spare L2 for adj)
// ---------------------------------------------------------------------------
__global__ void __launch_bounds__(256)
gat_softmax_row(const float* __restrict__ e1, const float* __restrict__ e2,
                const unsigned char* __restrict__ adj, float* __restrict__ out) {
    __shared__ float srow[N];          // 32 KB
    __shared__ float red[256];

    const int t   = threadIdx.x;
    const int bid = blockIdx.x;        // 0 .. B*N-1
    const int b   = bid >> 13;         // /N
    const int i   = bid & (N - 1);

    const float e1i = e1[(size_t)b * N + i];
    const float* __restrict__ e2b = e2 + (size_t)b * N;
    const unsigned char* __restrict__ arow = adj + (size_t)i * N;

    // ---- pass A: masked lrelu scores into LDS, track local max ----
    float lmax = -INFINITY;
    #pragma unroll
    for (int k = 0; k < N / 1024; ++k) {
        const int j = k * 1024 + t * 4;
        const unsigned aw = *(const unsigned*)(arow + j);     // 4 adjacency bytes
        v4f ev = *(const v4f*)(e2b + j);
        v4f sv;
        #pragma unroll
        for (int c = 0; c < 4; ++c) {
            float s = e1i + ev[c];
            s = (s >= 0.0f) ? s : SLOPE * s;
            const bool edge = (((aw >> (8 * c)) & 0xFFu) != 0u) | ((j + c) == i);
            s = edge ? s : -INFINITY;
            sv[c] = s;
            lmax = fmaxf(lmax, s);
        }
        *(v4f*)(srow + j) = sv;
    }

    // ---- block max reduction ----
    red[t] = lmax;
    __syncthreads();
    #pragma unroll
    for (int o = 128; o > 0; o >>= 1) {
        if (t < o) red[t] = fmaxf(red[t], red[t + o]);
        __syncthreads();
    }
    const float m = red[0];
    __syncthreads();

    // ---- pass B: exp(s - m) back into LDS, track local sum ----
    float lsum = 0.0f;
    #pragma unroll
    for (int k = 0; k < N / 1024; ++k) {
        const int j = k * 1024 + t * 4;
        v4f sv = *(const v4f*)(srow + j);
        #pragma unroll
        for (int c = 0; c < 4; ++c) {
            const float p = __expf(sv[c] - m);   // exp(-inf)=0 for masked
            sv[c] = p;
            lsum += p;
        }
        *(v4f*)(srow + j) = sv;
    }

    // ---- block sum reduction ----
    red[t] = lsum;
    __syncthreads();
    #pragma unroll
    for (int o = 128; o > 0; o >>= 1) {
        if (t < o) red[t] += red[t + o];
        __syncthreads();
    }
    const float inv = 1.0f / red[0];   // sum >= 1 (self-loop), safe
    __syncthreads();

    // ---- pass C: scaled streaming writeback (non-temporal b128 stores) ----
    float* __restrict__ orow = out + ((size_t)b * N + i) * N;
    #pragma unroll
    for (int k = 0; k < N / 1024; ++k) {
        const int j = k * 1024 + t * 4;
        v4f sv = *(const v4f*)(srow + j);
        sv[0] *= inv; sv[1] *= inv; sv[2] *= inv; sv[3] *= inv;
        __builtin_nontemporal_store(sv, (v4f*)(orow + j));
    }
}

extern "C" void kernel_launch(void* const* d_in, const int* in_sizes, int n_in,
                              void* d_out, int out_size, void* d_ws, size_t ws_size,
                              hipStream_t stream) {
    const float*         h   = (const float*)d_in[0];          // (B,F,N) f32
    const unsigned char* adj = (const unsigned char*)d_in[1];  // (N,N) bool (1 B/elem)
    const float*         W   = (const float*)d_in[2];          // (2,F) f32
    float* out = (float*)d_out;                                // (B,N,N) f32

    float* e1 = (float*)d_ws;            // B*N floats
    float* e2 = e1 + (size_t)B * N;      // B*N floats  (256 KB total in d_ws)

    // 2048 wave-tiles (4 batches * 512 col-tiles), 8 waves per 256-thread block
    gat_proj_wmma<<<(B * (N / 16)) / 8, 256, 0, stream>>>(h, W, e1, e2);

    // one block per (batch, row)
    gat_softmax_row<<<B * N, 256, 0, stream>>>(e1, e2, adj, out);
}